// MoETrajectoryBias_23545010716761
// MI455X (gfx1250) — compile-verified
//
#include <hip/hip_runtime.h>
#include <hip/hip_bf16.h>
#include <math.h>

// Problem constants (fixed by the reference).
#define E_  10
#define DIN 7
#define HIDN 56
#define NH  16
#define BB  2
#define SS  2048
#define TT  (BB * SS)          // 4096 tokens
#define NTILES (TT / 16)       // 256 token tiles

typedef float v2f __attribute__((ext_vector_type(2)));
typedef float v4f __attribute__((ext_vector_type(4)));
typedef float v8f __attribute__((ext_vector_type(8)));

__device__ __forceinline__ float gelu_erf(float x) {
    return 0.5f * x * (1.0f + erff(x * 0.70710678118654752440f));
}

// D = A(16x4) * B(4x16) + C, f32, wave32 WMMA.
// Lane layouts assumed (ISA 7.12.2):
//   A: lane L<16: {A[M=L][0], A[M=L][1]}, lane L>=16: {A[M=L-16][2], A[M=L-16][3]}
//   B: lane L<16: {B[0][N=L], B[1][N=L]}, lane L>=16: {B[2][N=L-16], B[3][N=L-16]}
//   C/D: VGPR r, lane L<16 -> (M=r, N=L); lane L>=16 -> (M=r+8, N=L-16)
__device__ __forceinline__ v8f wmma4(v2f a, v2f b, v8f c) {
#if defined(__gfx1250__) && __has_builtin(__builtin_amdgcn_wmma_f32_16x16x4_f32)
    return __builtin_amdgcn_wmma_f32_16x16x4_f32(
        /*neg_a=*/false, a, /*neg_b=*/false, b,
        /*c_mod=*/(short)0, c, /*reuse_a=*/false, /*reuse_b=*/false);
#else
    // Cross-lane emulation (wave32), same layout contract as above.
    int lane = (int)(threadIdx.x & 31u);
    int n = lane & 15;
    int hs = lane >> 4;
    float B0 = __shfl(b.x, n, 32);
    float B1 = __shfl(b.y, n, 32);
    float B2 = __shfl(b.x, n + 16, 32);
    float B3 = __shfl(b.y, n + 16, 32);
    v8f d;
    for (int r = 0; r < 8; ++r) {
        int M = r + 8 * hs;
        float A0 = __shfl(a.x, M, 32);
        float A1 = __shfl(a.y, M, 32);
        float A2 = __shfl(a.x, M + 16, 32);
        float A3 = __shfl(a.y, M + 16, 32);
        d[r] = c[r] + A0 * B0 + A1 * B1 + A2 * B2 + A3 * B3;
    }
    return d;
#endif
}

// ---------------------------------------------------------------------------
// Kernel 1: dense-over-experts MoE MLP via f32 WMMA, hard select -> pb (T,NH)
// One wave per 16-token tile; 4 waves / block; per-wave LDS slices.
// ---------------------------------------------------------------------------
__global__ __launch_bounds__(128) void moe_pb_kernel(
    const float* __restrict__ scalars,   // (T, DIN)
    const int*   __restrict__ ids,       // (T)
    const float* __restrict__ W1, const float* __restrict__ b1,   // (E,56,7),(E,56)
    const float* __restrict__ W2, const float* __restrict__ b2,   // (E,56,56),(E,56)
    const float* __restrict__ W3, const float* __restrict__ b3,   // (E,16,56),(E,16)
    float* __restrict__ pb)              // (T, NH)
{
    __shared__ __attribute__((aligned(16))) float h1buf[4][16][64];
    __shared__ __attribute__((aligned(16))) float h2buf[4][16][64];

    const int lane = (int)(threadIdx.x & 31u);
    const int w    = (int)(threadIdx.x >> 5u);
    const int m    = lane & 15;      // N / M-in-half index
    const int hs   = lane >> 4;      // half select
    const int tile = (int)blockIdx.x * 4 + w;
    const int t0   = tile * 16;

    // Preload A for layer 1 (K=7 padded to 8 -> 2 k-steps of 4).
    v2f a1[2];
    for (int kk = 0; kk < 2; ++kk) {
        int k0 = kk * 4 + hs * 2;
        float ax = (k0     < DIN) ? scalars[(t0 + m) * DIN + k0]     : 0.0f;
        float ay = (k0 + 1 < DIN) ? scalars[(t0 + m) * DIN + k0 + 1] : 0.0f;
        a1[kk].x = ax; a1[kk].y = ay;
    }
    // Expert ids of the 8 tokens this lane owns in C/D layout (M = r + 8*hs).
    int myids[8];
    for (int r = 0; r < 8; ++r) myids[r] = ids[t0 + r + 8 * hs];

    v8f sel = {0.f, 0.f, 0.f, 0.f, 0.f, 0.f, 0.f, 0.f};

    for (int e = 0; e < E_; ++e) {
        // ---- layer 1: (16x7) x (7x56->64) ----
        v8f c1[4];
        for (int nt = 0; nt < 4; ++nt) {
            int o = nt * 16 + m;
            float bv = (o < HIDN) ? b1[e * HIDN + o] : 0.0f;
            c1[nt] = (v8f){bv, bv, bv, bv, bv, bv, bv, bv};
        }
        for (int kk = 0; kk < 2; ++kk) {
            int k0 = kk * 4 + hs * 2;
            for (int nt = 0; nt < 4; ++nt) {
                int o = nt * 16 + m;
                v2f bm;
                bm.x = (o < HIDN && k0     < DIN) ? W1[(e * HIDN + o) * DIN + k0]     : 0.0f;
                bm.y = (o < HIDN && k0 + 1 < DIN) ? W1[(e * HIDN + o) * DIN + k0 + 1] : 0.0f;
                c1[nt] = wmma4(a1[kk], bm, c1[nt]);
            }
        }
        for (int nt = 0; nt < 4; ++nt)
            for (int r = 0; r < 8; ++r)
                h1buf[w][r + 8 * hs][nt * 16 + m] = gelu_erf(c1[nt][r]);
        __syncthreads();

        // ---- layer 2: (16x56) x (56x56->64), K = 56 = 14 x 4 ----
        v8f c2[4];
        for (int nt = 0; nt < 4; ++nt) {
            int o = nt * 16 + m;
            float bv = (o < HIDN) ? b2[e * HIDN + o] : 0.0f;
            c2[nt] = (v8f){bv, bv, bv, bv, bv, bv, bv, bv};
        }
        for (int kk = 0; kk < 14; ++kk) {
            int k0 = kk * 4 + hs * 2;
            v2f a = *reinterpret_cast<const v2f*>(&h1buf[w][m][k0]);
            for (int nt = 0; nt < 4; ++nt) {
                int o = nt * 16 + m;
                v2f bm;
                bm.x = (o < HIDN) ? W2[(e * HIDN + o) * HIDN + k0]     : 0.0f;
                bm.y = (o < HIDN) ? W2[(e * HIDN + o) * HIDN + k0 + 1] : 0.0f;
                c2[nt] = wmma4(a, bm, c2[nt]);
            }
        }
        for (int nt = 0; nt < 4; ++nt)
            for (int r = 0; r < 8; ++r)
                h2buf[w][r + 8 * hs][nt * 16 + m] = gelu_erf(c2[nt][r]);
        __syncthreads();

        // ---- layer 3: (16x56) x (56x16), single N tile ----
        float bv3 = b3[e * NH + m];
        v8f c3 = (v8f){bv3, bv3, bv3, bv3, bv3, bv3, bv3, bv3};
        for (int kk = 0; kk < 14; ++kk) {
            int k0 = kk * 4 + hs * 2;
            v2f a = *reinterpret_cast<const v2f*>(&h2buf[w][m][k0]);
            v2f bm;
            bm.x = W3[(e * NH + m) * HIDN + k0];
            bm.y = W3[(e * NH + m) * HIDN + k0 + 1];
            c3 = wmma4(a, bm, c3);
        }
        for (int r = 0; r < 8; ++r)
            sel[r] = (myids[r] == e) ? c3[r] : sel[r];
        __syncthreads();  // protect h1buf/h2buf reads vs next iteration's stores
    }

    // pb[t][h]: t = t0 + r + 8*hs, h = m
    for (int r = 0; r < 8; ++r)
        pb[(t0 + r + 8 * hs) * NH + m] = sel[r];
}

// ---------------------------------------------------------------------------
// Kernel 2: per-(b,h) mean of gathered scales/offsets, then the 2048-entry
// exp distance table per (b,h):  ktab[b][h][d] = exp(-avg_s*0.01*d + avg_o)
// ---------------------------------------------------------------------------
__global__ __launch_bounds__(256) void reduce_ktab_kernel(
    const int*   __restrict__ ids,     // (T)
    const float* __restrict__ dscale,  // (E, NH)
    const float* __restrict__ doff,    // (E, NH)
    float* __restrict__ ktab)          // (B, NH, SS)
{
    __shared__ float s_s[256], s_o[256];
    __shared__ float avs[NH], avo[NH];
    const int b   = (int)blockIdx.x;
    const int tid = (int)threadIdx.x;
    const int h   = tid & 15;
    const int g   = tid >> 4;

    float ss = 0.0f, so = 0.0f;
    for (int s = g; s < SS; s += 16) {
        int e = ids[b * SS + s];
        ss += dscale[e * NH + h];
        so += doff[e * NH + h];
    }
    s_s[tid] = ss; s_o[tid] = so;
    __syncthreads();
    if (tid < NH) {
        float a = 0.0f, o = 0.0f;
        for (int gg = 0; gg < 16; ++gg) { a += s_s[tid + (gg << 4)]; o += s_o[tid + (gg << 4)]; }
        avs[tid] = a / (float)SS;
        avo[tid] = o / (float)SS;
    }
    __syncthreads();
    for (int idx = tid; idx < NH * SS; idx += 256) {
        int hh = idx >> 11;        // / SS
        int d  = idx & (SS - 1);
        ktab[b * NH * SS + idx] = expf(-avs[hh] * 0.01f * (float)d + avo[hh]);
    }
}

// ---------------------------------------------------------------------------
// Kernel 3: out[b,h,i,j] = pb[b,i,h] * ktab[b,h,|i-j|].  512 MB of streaming
// nontemporal stores; table row cached in LDS and reused over 32 rows.
// grid = B*NH*(SS/32) = 2048 blocks.
// ---------------------------------------------------------------------------
__global__ __launch_bounds__(256) void expand_kernel(
    const float* __restrict__ pb,    // (T, NH)
    const float* __restrict__ ktab,  // (B, NH, SS)
    float* __restrict__ out)         // (B, NH, SS, SS)
{
    __shared__ float kl[SS];   // 8 KB
    __shared__ float pl[32];
    const int blk   = (int)blockIdx.x;
    const int chunk = blk & 63;          // SS/32 chunks of rows
    const int bh    = blk >> 6;          // b*NH + h
    const int b     = bh >> 4;
    const int h     = bh & 15;
    const int tid   = (int)threadIdx.x;

    const float* kt = ktab + (size_t)bh * SS;
    for (int q = tid; q < SS; q += 256) kl[q] = kt[q];
    const int i0 = chunk * 32;
    if (tid < 32) pl[tid] = pb[(b * SS + i0 + tid) * NH + h];
    __syncthreads();

    float* obase = out + (size_t)bh * SS * SS;
    for (int r = 0; r < 32; ++r) {
        const int i = i0 + r;
        const float pv = pl[r];
        float* orow = obase + (size_t)i * SS;
        for (int j0 = tid * 4; j0 < SS; j0 += 1024) {
            int d0 = i - j0;       d0 = d0 < 0 ? -d0 : d0;
            int d1 = i - (j0 + 1); d1 = d1 < 0 ? -d1 : d1;
            int d2 = i - (j0 + 2); d2 = d2 < 0 ? -d2 : d2;
            int d3 = i - (j0 + 3); d3 = d3 < 0 ? -d3 : d3;
            v4f v;
            v.x = pv * kl[d0];
            v.y = pv * kl[d1];
            v.z = pv * kl[d2];
            v.w = pv * kl[d3];
            __builtin_nontemporal_store(v, reinterpret_cast<v4f*>(orow + j0));
        }
    }
}

extern "C" void kernel_launch(void* const* d_in, const int* in_sizes, int n_in,
                              void* d_out, int out_size, void* d_ws, size_t ws_size,
                              hipStream_t stream) {
    (void)in_sizes; (void)n_in; (void)out_size; (void)ws_size;
    const float* scalars = (const float*)d_in[0];  // (B,S,DIN)
    const int*   ids     = (const int*)  d_in[1];  // (B,S)
    //           seq_len =               d_in[2]   // unused (S fixed)
    const float* W1      = (const float*)d_in[3];
    const float* b1      = (const float*)d_in[4];
    const float* W2      = (const float*)d_in[5];
    const float* b2      = (const float*)d_in[6];
    const float* W3      = (const float*)d_in[7];
    const float* b3      = (const float*)d_in[8];
    const float* dscale  = (const float*)d_in[9];
    const float* doff    = (const float*)d_in[10];
    float* out = (float*)d_out;

    float* ws_pb   = (float*)d_ws;                   // T*NH floats   (256 KB)
    float* ws_ktab = ws_pb + (size_t)TT * NH;        // B*NH*SS floats (256 KB)

    moe_pb_kernel<<<NTILES / 4, 128, 0, stream>>>(scalars, ids, W1, b1, W2, b2,
                                                  W3, b3, ws_pb);
    reduce_ktab_kernel<<<BB, 256, 0, stream>>>(ids, dscale, doff, ws_ktab);
    expand_kernel<<<BB * NH * (SS / 32), 256, 0, stream>>>(ws_pb, ws_ktab, out);
}